// VectorQuantizerEMA_21036749816288
// MI455X (gfx1250) — compile-verified
//
#include <hip/hip_runtime.h>
#include <hip/hip_bf16.h>

#define K_CODES 512
#define DIM 64
#define T_LEN 4096
#define B_SZ 32
#define N_ROWS (B_SZ * T_LEN)       // 131072
#define NBLK (N_ROWS / 16)          // 8192 row-blocks of 16 rows
#define BLK_PER_WG 16
#define GRID_MAIN (NBLK / BLK_PER_WG)  // 512 workgroups

typedef float v2f __attribute__((ext_vector_type(2)));
typedef float v8f __attribute__((ext_vector_type(8)));

// ws layout (floats): [0..511] h = 0.5*||e_k||^2 ; [512..1023] uint counts ; [1024] lossSum

__global__ void vq_prep(const float* __restrict__ emb, float* __restrict__ ws) {
    int k = blockIdx.x * blockDim.x + threadIdx.x;
    if (k < K_CODES) {
        const float* e = emb + k * DIM;
        float s = 0.f;
#pragma unroll
        for (int d = 0; d < DIM; ++d) s += e[d] * e[d];
        ws[k] = 0.5f * s;
        ((unsigned int*)ws)[512 + k] = 0u;
    }
    if (k == 0) ws[1024] = 0.f;
}

__global__ __launch_bounds__(256) void vq_main(const float* __restrict__ x,
                                               const float* __restrict__ emb,
                                               float* __restrict__ out,
                                               float* __restrict__ ws) {
    const int tid  = threadIdx.x;
    const int w    = tid >> 5;       // wave 0..7 -> codes [w*64, w*64+64)
    const int lane = tid & 31;
    const int g    = lane >> 4;      // lane-half: selects K pair / row-half
    const int n    = lane & 15;      // code-in-tile / row-in-block

    __shared__ float lv[8 * 16];
    __shared__ int   li[8 * 16];
    __shared__ int   fIdx[16];
    __shared__ float lc[16];

    // Preload this wave's 64x64 codebook slice into registers (block-invariant).
    // B layout (4x16 f32): VGPR j, lane-half g -> K = 2g + j, N = lane&15.
    v2f   Breg[4][16];
    float hreg[4];
#pragma unroll
    for (int c = 0; c < 4; ++c) {
        const int code = w * 64 + c * 16 + n;
        const float* er = emb + code * DIM + 2 * g;
#pragma unroll
        for (int s = 0; s < 16; ++s)
            Breg[c][s] = *(const v2f*)(er + 4 * s);   // E[code][4s+2g .. +1]
        hreg[c] = ws[code];
    }

    unsigned int* counts  = (unsigned int*)ws + 512;
    float*        lossSum = ws + 1024;
    float*        qout    = out + 1;
    float*        idxOut  = out + 1 + (size_t)B_SZ * DIM * T_LEN + 1;

    for (int it = 0; it < BLK_PER_WG; ++it) {
        const int blk = blockIdx.x * BLK_PER_WG + it;
        const int n0  = blk * 16;
        const int b   = n0 >> 12;        // T = 4096
        const int t0  = n0 & 4095;
        const float* xb = x + (size_t)b * DIM * T_LEN + t0 + n;

        // A tile: 16 rows x 64 dims. A layout (16x4 f32): lane-half g -> K = 2g+j, M = lane&15.
        // x_flat[row][d] = inputs[b, d, t0+row]  (stride T along d)
        v2f Areg[16];
#pragma unroll
        for (int s = 0; s < 16; ++s) {
            const int d0 = 4 * s + 2 * g;
            Areg[s].x = xb[(size_t)d0 * T_LEN];
            Areg[s].y = xb[(size_t)(d0 + 1) * T_LEN];
        }

        v8f acc[4] = {};
#pragma unroll
        for (int s = 0; s < 16; ++s) {
#pragma unroll
            for (int c = 0; c < 4; ++c) {
                acc[c] = __builtin_amdgcn_wmma_f32_16x16x4_f32(
                    false, Areg[s], false, Breg[c][s], (short)0, acc[c], false, false);
            }
        }

        // Per-lane argmax of score = G - 0.5||e||^2 over this wave's 4 col-tiles.
        // C layout: VGPR r, lane-half g -> row = r + 8g, col = lane&15.
        float bestV[8]; int bestI[8];
#pragma unroll
        for (int r = 0; r < 8; ++r) {
            float v = acc[0][r] - hreg[0];
            int   i = w * 64 + n;
#pragma unroll
            for (int c = 1; c < 4; ++c) {
                const float u = acc[c][r] - hreg[c];
                const int   j = w * 64 + c * 16 + n;
                if (u > v) { v = u; i = j; }   // strict > keeps first index on ties
            }
            bestV[r] = v; bestI[r] = i;
        }
        // Reduce across the 16 lanes of each half (codes n=0..15), index tiebreak.
#pragma unroll
        for (int mask = 1; mask <= 8; mask <<= 1) {
#pragma unroll
            for (int r = 0; r < 8; ++r) {
                const float ov = __shfl_xor(bestV[r], mask, 32);
                const int   oi = __shfl_xor(bestI[r], mask, 32);
                if (ov > bestV[r] || (ov == bestV[r] && oi < bestI[r])) {
                    bestV[r] = ov; bestI[r] = oi;
                }
            }
        }

        // Row norms ||x_row||^2 from the A registers (used for the commitment loss).
        float part = 0.f;
#pragma unroll
        for (int s = 0; s < 16; ++s)
            part += Areg[s].x * Areg[s].x + Areg[s].y * Areg[s].y;
        const float rnorm = part + __shfl_xor(part, 16, 32);  // lane m holds row m's norm

        if (n == 0) {
#pragma unroll
            for (int r = 0; r < 8; ++r) {
                lv[w * 16 + r + 8 * g] = bestV[r];
                li[w * 16 + r + 8 * g] = bestI[r];
            }
        }
        __syncthreads();

        if (tid < 16) {   // wave 0 lanes 0..15: row = tid, rnorm register matches row
            const int row = tid;
            float v = lv[row]; int i = li[row];
#pragma unroll
            for (int w2 = 1; w2 < 8; ++w2) {   // ascending code ranges: strict > = first tie
                const float u = lv[w2 * 16 + row];
                if (u > v) { v = u; i = li[w2 * 16 + row]; }
            }
            fIdx[row] = i;
            idxOut[n0 + row] = (float)i;
            atomicAdd(&counts[i], 1u);
            // ||x - e||^2 = ||x||^2 - 2*(g - 0.5||e||^2) = rnorm - 2*v
            lc[row] = rnorm - 2.0f * v;
        }
        __syncthreads();

        if (tid == 0) {
            float s = 0.f;
#pragma unroll
            for (int r = 0; r < 16; ++r) s += lc[r];
            atomicAdd(lossSum, s);
        }

        // quantized_ste forward value == E[idx]; write back in [B, D, T] layout.
        {
            const int d  = tid >> 2;
            const int mq = (tid & 3) * 4;
            float4 qv;
            qv.x = emb[fIdx[mq + 0] * DIM + d];
            qv.y = emb[fIdx[mq + 1] * DIM + d];
            qv.z = emb[fIdx[mq + 2] * DIM + d];
            qv.w = emb[fIdx[mq + 3] * DIM + d];
            *(float4*)(qout + (size_t)b * DIM * T_LEN + (size_t)d * T_LEN + t0 + mq) = qv;
        }
    }
}

__global__ void vq_final(const float* __restrict__ ws, float* __restrict__ out) {
    __shared__ float sh[256];
    const int tid = threadIdx.x;
    const unsigned int* counts = (const unsigned int*)ws + 512;
    float s = 0.f;
    for (int k = tid; k < K_CODES; k += 256) {
        const float p = (float)counts[k] / (float)N_ROWS;
        s += p * logf(p + 1e-10f);
    }
    sh[tid] = s;
    __syncthreads();
    for (int off = 128; off > 0; off >>= 1) {
        if (tid < off) sh[tid] += sh[tid + off];
        __syncthreads();
    }
    if (tid == 0) {
        out[1 + (size_t)B_SZ * DIM * T_LEN] = expf(-sh[0]);                 // perplexity
        out[0] = 0.25f * ws[1024] / (float)((size_t)N_ROWS * DIM);          // loss
    }
}

extern "C" void kernel_launch(void* const* d_in, const int* in_sizes, int n_in,
                              void* d_out, int out_size, void* d_ws, size_t ws_size,
                              hipStream_t stream) {
    const float* x   = (const float*)d_in[0];   // [32, 64, 4096]
    const float* emb = (const float*)d_in[1];   // [512, 64]
    float* out = (float*)d_out;
    float* ws  = (float*)d_ws;

    vq_prep<<<2, 256, 0, stream>>>(emb, ws);
    vq_main<<<GRID_MAIN, 256, 0, stream>>>(x, emb, out, ws);
    vq_final<<<1, 256, 0, stream>>>(ws, out);
}